// GraclusMaxPool_49237505081500
// MI455X (gfx1250) — compile-verified
//
#include <hip/hip_runtime.h>
#include <math.h>

// ---------------------------------------------------------------------------
// GraclusMaxPool for MI455X (gfx1250, wave32).
// Bandwidth-bound pipeline (~115 MB total traffic, ~5 us at 23.3 TB/s) plus a
// serial graclus matching chain (N=100000 dependent steps, one wave).
// No matmul content -> CDNA5 path used: async global->LDS DMA + asynccnt,
// global prefetch, wave32 cross-lane ops.
// ---------------------------------------------------------------------------

#define WAVE 32
#define WPB  8          // waves per block in the feature-pool kernel

// Probe-confirmed (round 1): HAVE global_load_async_to_lds_b128,
// s_wait_asynccnt, tensor_load_to_lds, cluster_id_x.
typedef int v4i __attribute__((vector_size(16)));
typedef v4i __attribute__((address_space(1))) gv4i;
typedef v4i __attribute__((address_space(3))) lv4i;

// 16B per-lane async copy global -> LDS (CDNA5 GLOBAL_LOAD_ASYNC_TO_LDS_B128)
__device__ __forceinline__ void async_copy16(void* lds_dst, const void* gsrc) {
  __builtin_amdgcn_global_load_async_to_lds_b128(
      (gv4i*)(void*)gsrc,          // A-S1 source (const dropped; builtin is non-const)
      (lv4i*)lds_dst,              // A-S3 destination
      /*offset=*/0, /*cpol=*/0);
}

__device__ __forceinline__ void wait_async0() {
  __builtin_amdgcn_s_wait_asynccnt(0);
}

// ---------------------------------------------------------------------------
// 1) init workspace
__global__ void gmp_init_kernel(int* __restrict__ deg, int* __restrict__ cluster,
                                int* __restrict__ partner, int* __restrict__ maxv,
                                int n) {
  int i = blockIdx.x * blockDim.x + threadIdx.x;
  if (i < n) { deg[i] = 0; cluster[i] = -1; partner[i] = -1; }
  if (i == 0) *maxv = 0;  // bits of +0.0f
}

// 2) deg = segment_sum(ones, col)
__global__ void gmp_deg_kernel(const int* __restrict__ col, int* __restrict__ deg,
                               int e) {
  int i = blockIdx.x * blockDim.x + threadIdx.x;
  if (i < e) atomicAdd(&deg[col[i]], 1);
}

// 3) edge weights (normalized_cut_2d)
__global__ void gmp_weight_kernel(const int* __restrict__ row, const int* __restrict__ col,
                                  const float* __restrict__ pos, const int* __restrict__ deg,
                                  float* __restrict__ w, int e) {
  int i = blockIdx.x * blockDim.x + threadIdx.x;
  if (i >= e) return;
  int r = row[i], c = col[i];
  float dx = pos[2 * r] - pos[2 * c];
  float dy = pos[2 * r + 1] - pos[2 * c + 1];
  float dist = sqrtf(dx * dx + dy * dy);
  float di = 1.0f / ((float)deg[r] + 1e-8f) + 1.0f / ((float)deg[c] + 1e-8f);
  w[i] = dist * di;
}

// 4) serial graclus greedy matching: one wave, lanes 0..D-1 handle neighbors.
//    Matches jnp argmax semantics (first index wins ties).
__global__ __launch_bounds__(WAVE) void gmp_graclus_kernel(
    const int* __restrict__ col, const float* __restrict__ w,
    volatile int* cluster, int* __restrict__ partner, int n, int d) {
  int lane = threadIdx.x;
  for (int i = 0; i < n; ++i) {
    long long base = (long long)i * d;
    // prefetch next node's neighbor row (speculative, OOB-safe)
    __builtin_prefetch(col + base + d + lane, 0, 0);
    __builtin_prefetch(w + base + d + lane, 0, 0);

    int nb = -1; float wj = -__builtin_inff(); bool ok = false;
    if (lane < d) {
      nb = col[base + lane];
      wj = w[base + lane];
      ok = (cluster[nb] < 0) && (nb != i);
    }
    float bw = ok ? wj : -__builtin_inff();
    int   bj = ok ? lane : 0x7fffffff;   // tie-break: lowest neighbor slot
    int   bn = nb;
    #pragma unroll
    for (int off = 16; off; off >>= 1) {
      float ow = __shfl_xor(bw, off, WAVE);
      int   oj = __shfl_xor(bj, off, WAVE);
      int   on = __shfl_xor(bn, off, WAVE);
      if (ow > bw || (ow == bw && oj < bj)) { bw = ow; bj = oj; bn = on; }
    }
    bool has = (__ballot(ok) != 0ull);
    int ci = cluster[i];
    if (lane == 0 && ci < 0) {
      cluster[i] = i;                       // become a representative
      if (has) { cluster[bn] = i; partner[i] = bn; }
    }
    // same-wave store->load same-address ordering is architectural (ISA 7.3)
  }
}

// 5) feature max-pool. Cluster = {c} or {c, partner}: gather <=2 rows via
//    CDNA5 async DMA into LDS, elementwise max, coalesced 512B store.
//    Assumes F == 128 (32 lanes x float4).
__global__ __launch_bounds__(WPB * WAVE) void gmp_xpool_kernel(
    const float* __restrict__ x, const int* __restrict__ cluster,
    const int* __restrict__ partner, float* __restrict__ xout, int n, int f) {
  __shared__ __align__(16) float lds[WPB * 2 * 128];
  int wave = threadIdx.x >> 5;
  int lane = threadIdx.x & 31;
  int c = blockIdx.x * WPB + wave;
  if (c >= n) return;                       // wave-uniform

  float4* dst = (float4*)(xout + (size_t)c * f);
  if (cluster[c] != c) {                    // unoccupied segment -> zeros
    dst[lane] = make_float4(0.f, 0.f, 0.f, 0.f);
    return;
  }
  int p = partner[c];
  float* lc = &lds[wave * 2 * 128];
  const float4* rowc = (const float4*)(x + (size_t)c * f);
  async_copy16(&lc[lane * 4], &rowc[lane]);
  if (p >= 0) {
    const float4* rowp = (const float4*)(x + (size_t)p * f);
    async_copy16(&lc[128 + lane * 4], &rowp[lane]);
  }
  wait_async0();
  float4 a = ((const float4*)lc)[lane];
  if (p >= 0) {
    float4 b = ((const float4*)(lc + 128))[lane];
    a.x = fmaxf(a.x, b.x); a.y = fmaxf(a.y, b.y);
    a.z = fmaxf(a.z, b.z); a.w = fmaxf(a.w, b.w);
  }
  dst[lane] = a;
}

// 6) position mean-pool (count is 1 or 2)
__global__ void gmp_pospool_kernel(const float* __restrict__ pos,
                                   const int* __restrict__ cluster,
                                   const int* __restrict__ partner,
                                   float* __restrict__ pos_out, int n) {
  int c = blockIdx.x * blockDim.x + threadIdx.x;
  if (c >= n) return;
  float ox = 0.f, oy = 0.f;
  if (cluster[c] == c) {
    int p = partner[c];
    ox = pos[2 * c]; oy = pos[2 * c + 1];
    if (p >= 0) {
      ox = (ox + pos[2 * p]) * 0.5f;
      oy = (oy + pos[2 * p + 1]) * 0.5f;
    }
  }
  pos_out[2 * c] = ox; pos_out[2 * c + 1] = oy;
}

// 7) global max |cart| over non-self pooled edges (nonneg-float-bits atomicMax)
__global__ void gmp_maxv_kernel(const int* __restrict__ row, const int* __restrict__ col,
                                const int* __restrict__ cluster,
                                const float* __restrict__ pos_out,
                                int* __restrict__ maxv, int e) {
  int i = blockIdx.x * blockDim.x + threadIdx.x;
  float m = 0.0f;
  if (i < e) {
    int rc = cluster[row[i]], cc = cluster[col[i]];
    if (rc != cc) {
      float cx = fabsf(pos_out[2 * cc] - pos_out[2 * rc]);
      float cy = fabsf(pos_out[2 * cc + 1] - pos_out[2 * rc + 1]);
      m = fmaxf(cx, cy);
    }
  }
  #pragma unroll
  for (int off = 16; off; off >>= 1) m = fmaxf(m, __shfl_xor(m, off, WAVE));
  if ((threadIdx.x & 31) == 0 && m > 0.0f)
    atomicMax(maxv, __float_as_int(m));   // valid: all candidates >= 0
}

// 8) pooled edge_index (as float values) + Cartesian attrs + self loops
__global__ void gmp_attr_kernel(const int* __restrict__ row, const int* __restrict__ col,
                                const int* __restrict__ cluster,
                                const float* __restrict__ pos_out,
                                const int* __restrict__ maxv,
                                float* __restrict__ ei_out, float* __restrict__ ea_out,
                                int e, int n) {
  int i = blockIdx.x * blockDim.x + threadIdx.x;
  int total = e + n;
  if (i >= total) return;
  if (i < e) {
    int rc = cluster[row[i]], cc = cluster[col[i]];
    ei_out[i] = (float)rc;
    ei_out[total + i] = (float)cc;
    float ax = 1.0f, ay = 1.0f;
    if (rc != cc) {
      float mv2 = 2.0f * __int_as_float(*maxv);
      ax = (pos_out[2 * cc] - pos_out[2 * rc]) / mv2 + 0.5f;
      ay = (pos_out[2 * cc + 1] - pos_out[2 * rc + 1]) / mv2 + 0.5f;
    }
    ea_out[2 * i] = ax; ea_out[2 * i + 1] = ay;
  } else {
    int s = i - e;                       // appended self loop
    ei_out[i] = (float)s;
    ei_out[total + i] = (float)s;
    ea_out[2 * i] = 1.0f; ea_out[2 * i + 1] = 1.0f;
  }
}

// ---------------------------------------------------------------------------
extern "C" void kernel_launch(void* const* d_in, const int* in_sizes, int n_in,
                              void* d_out, int out_size, void* d_ws, size_t ws_size,
                              hipStream_t stream) {
  const float* x   = (const float*)d_in[0];
  const float* pos = (const float*)d_in[1];
  const int*   ei  = (const int*)d_in[2];

  const int n = in_sizes[1] / 2;        // 100000
  const int e = in_sizes[2] / 2;        // 1600000
  const int f = in_sizes[0] / n;        // 128
  const int d = e / n;                  // 16
  const int* row = ei;
  const int* col = ei + e;

  // workspace carve-out (256B aligned): deg, cluster, partner, w, maxv
  char* ws = (char*)d_ws;
  auto carve = [&](size_t bytes) {
    char* p = ws;
    ws += (bytes + 255) & ~(size_t)255;
    return p;
  };
  int*   deg     = (int*)carve((size_t)n * sizeof(int));
  int*   cluster = (int*)carve((size_t)n * sizeof(int));
  int*   partner = (int*)carve((size_t)n * sizeof(int));
  float* w       = (float*)carve((size_t)e * sizeof(float));
  int*   maxv    = (int*)carve(sizeof(int));
  (void)ws_size;

  // output layout: x_out[N*F] | pos_out[N*2] | edge_index_out[2*(E+N)] | edge_attr[(E+N)*2]
  float* xo  = (float*)d_out;
  float* po  = xo + (size_t)n * f;
  float* eio = po + 2 * (size_t)n;
  float* eao = eio + 2 * ((size_t)e + n);
  (void)out_size;

  const int B = 256;
  gmp_init_kernel<<<(n + B - 1) / B, B, 0, stream>>>(deg, cluster, partner, maxv, n);
  gmp_deg_kernel<<<(e + B - 1) / B, B, 0, stream>>>(col, deg, e);
  gmp_weight_kernel<<<(e + B - 1) / B, B, 0, stream>>>(row, col, pos, deg, w, e);
  gmp_graclus_kernel<<<1, WAVE, 0, stream>>>(col, w, (volatile int*)cluster, partner, n, d);
  gmp_xpool_kernel<<<(n + WPB - 1) / WPB, WPB * WAVE, 0, stream>>>(x, cluster, partner, xo, n, f);
  gmp_pospool_kernel<<<(n + B - 1) / B, B, 0, stream>>>(pos, cluster, partner, po, n);
  gmp_maxv_kernel<<<(e + B - 1) / B, B, 0, stream>>>(row, col, cluster, po, maxv, e);
  gmp_attr_kernel<<<(e + n + B - 1) / B, B, 0, stream>>>(row, col, cluster, po, maxv, eio, eao, e, n);
}